// MultiHeadAttentionCausalMasked_47064251629941
// MI455X (gfx1250) — compile-verified
//
#include <hip/hip_runtime.h>
#include <hip/hip_bf16.h>
#include <cstdint>

// ---------------------------------------------------------------------------
// Causal linear attention (Katharopoulos) for B=1, S=768, DM=512, H=8, D=64.
//   out[l,h,e] = z[l,h] * sum_{j<=l} (phi_q[l,h,:].phi_k[j,h,:]) * v[j,h,e]
//   z[l,h]     = 1 / (phi_q[l,h,:] . sum_j phi_k[j,h,:])   (full key sum)
// All GEMMs use V_WMMA_F32_16X16X4_F32 (full fp32 precision on matrix pipe).
// Global->LDS staging uses GLOBAL_LOAD_ASYNC_TO_LDS (ASYNCcnt) when available.
// ---------------------------------------------------------------------------

#define S_LEN  768
#define DMODEL 512
#define NHEAD  8
#define DHEAD  64

#define AS1 __attribute__((address_space(1)))
#define AS3 __attribute__((address_space(3)))

#if defined(__has_builtin)
#if __has_builtin(__builtin_amdgcn_global_load_async_to_lds_b128) && \
    __has_builtin(__builtin_amdgcn_global_load_async_to_lds_b32) &&  \
    __has_builtin(__builtin_amdgcn_s_wait_asynccnt)
#define HAVE_ASYNC_LDS 1
#endif
#endif
#ifndef HAVE_ASYNC_LDS
#define HAVE_ASYNC_LDS 0
#endif

typedef __attribute__((ext_vector_type(2))) float v2f;
typedef __attribute__((ext_vector_type(8))) float v8f;
typedef __attribute__((ext_vector_type(4))) int v4i;

// Global (AS1) / LDS (AS3) typed pointer casts via integer round-trip:
// flat global == AS1 numerically; flat LDS pointer's low 32 bits are the LDS
// offset, which is what an AS3 pointer holds.
#define GPTR(T, p) ((AS1 T*)(uintptr_t)(p))
#define LPTR(T, p) ((AS3 T*)(uintptr_t)(p))

static __device__ __forceinline__ int lane_id() { return (int)(threadIdx.x & 31u); }

static __device__ __forceinline__ v8f wmma_f32(v2f a, v2f b, v8f c) {
  // D = A(16x4,f32) * B(4x16,f32) + C(16x16,f32); 8-arg VOP3P form.
  return __builtin_amdgcn_wmma_f32_16x16x4_f32(
      /*neg_a=*/false, a, /*neg_b=*/false, b,
      /*c_mod=*/(short)0, c, /*reuse_a=*/false, /*reuse_b=*/false);
}

// Fragment where element (m, k) = lds[(row0+m)*stride + k]   (row-major tile).
static __device__ __forceinline__ v2f frag_rowmajor(const float* lds, int row0,
                                                    int k0, int stride) {
  const int l = lane_id();
  const int r = row0 + (l & 15);
  const int kk = k0 + ((l >> 4) << 1);        // lanes 16-31 hold K=+2,+3
  const float* p = lds + r * stride + kk;
  v2f f; f.x = p[0]; f.y = p[1];
  return f;
}

// Fragment where element (m, k) = lds[k*stride + m]          (k-major tile).
static __device__ __forceinline__ v2f frag_colmajor(const float* lds, int m0,
                                                    int k0, int stride) {
  const int l = lane_id();
  const int m = m0 + (l & 15);
  const int kk = k0 + ((l >> 4) << 1);
  v2f f; f.x = lds[kk * stride + m]; f.y = lds[(kk + 1) * stride + m];
  return f;
}

// 16x16 f32 C/D layout: VGPR i -> M = i (lanes 0-15) / i+8 (lanes 16-31).
static __device__ __forceinline__ void acc_rc(int& rbase, int& c) {
  const int l = lane_id();
  c = l & 15;
  rbase = (l >> 4) << 3;
}

// ---------------------------------------------------------------------------
// Kernel 1: Y = phi?(X @ W + b), X:(768,512) W:(512,512).
// Workgroup tile 128x64, 8 waves, each wave a 16x64 strip (4 accumulators).
// Inner loop is software-pipelined one k-step deep so DS fragment loads for
// k+4 are in flight while the WMMAs for k execute.
// ---------------------------------------------------------------------------
#define K1_TM 128
#define K1_TN 64
#define K1_TK 64
#define K1_LDA 68   // pad: float4-aligned staging, conflict-free frag reads
#define K1_LDB 68

__global__ __launch_bounds__(256) void proj_phi_kernel(
    const float* __restrict__ X, const float* __restrict__ W,
    const float* __restrict__ bias, float* __restrict__ Y, int apply_phi) {
  __shared__ float As[K1_TM * K1_LDA];   // X tile, row-major [m][k]
  __shared__ float Bs[K1_TN * K1_LDB];   // W tile, transposed [n][k]

  const int m0 = blockIdx.y * K1_TM;
  const int n0 = blockIdx.x * K1_TN;
  const int tid = (int)threadIdx.x;
  const int wave = tid >> 5;
  const int wrow = wave * 16;

  v8f acc[4];
#pragma unroll
  for (int n = 0; n < 4; ++n) acc[n] = v8f{0, 0, 0, 0, 0, 0, 0, 0};

  for (int kt = 0; kt < DMODEL; kt += K1_TK) {
    __syncthreads();
    // ---- Stage A tile: 128x64 floats (16B-granule copies). ----
#pragma unroll
    for (int j = 0; j < 8; ++j) {
      const int idx = tid + j * 256;            // 0..2047 float4s
      const int row = idx >> 4;
      const int c4 = idx & 15;
      const float* src = &X[(m0 + row) * DMODEL + kt + c4 * 4];
      float* dst = &As[row * K1_LDA + c4 * 4];
#if HAVE_ASYNC_LDS
      __builtin_amdgcn_global_load_async_to_lds_b128(
          GPTR(v4i, src), LPTR(v4i, dst), 0, 0);
#else
      *(float4*)dst = *(const float4*)src;
#endif
    }
    // ---- Stage B tile transposed: Bs[n][k] = W[kt+k][n0+n]. ----
#if HAVE_ASYNC_LDS
#pragma unroll
    for (int j = 0; j < 16; ++j) {
      const int idx = tid + j * 256;            // 0..4095
      const int kk = idx >> 6;
      const int nn = idx & 63;
      __builtin_amdgcn_global_load_async_to_lds_b32(
          GPTR(int, &W[(kt + kk) * DMODEL + n0 + nn]),
          LPTR(int, &Bs[nn * K1_LDB + kk]), 0, 0);
    }
#else
#pragma unroll
    for (int j = 0; j < 4; ++j) {
      const int idx = tid + j * 256;            // 0..1023 float4s
      const int kk = idx >> 4;
      const int n4 = (idx & 15) * 4;
      const float4 wv = *(const float4*)&W[(kt + kk) * DMODEL + n0 + n4];
      Bs[(n4 + 0) * K1_LDB + kk] = wv.x;
      Bs[(n4 + 1) * K1_LDB + kk] = wv.y;
      Bs[(n4 + 2) * K1_LDB + kk] = wv.z;
      Bs[(n4 + 3) * K1_LDB + kk] = wv.w;
    }
#endif
    // Prefetch next K-tile into GL2 while this one computes.
    if (kt + K1_TK < DMODEL) {
      const int prow = tid >> 1;
      const int phalf = (tid & 1) * 32;
      __builtin_prefetch(&X[(m0 + prow) * DMODEL + kt + K1_TK + phalf], 0, 1);
      const int pk = tid >> 2;
      const int pn = (tid & 3) * 16;
      __builtin_prefetch(&W[(kt + K1_TK + pk) * DMODEL + n0 + pn], 0, 1);
    }
#if HAVE_ASYNC_LDS
    __builtin_amdgcn_s_wait_asynccnt(0);   // drain this wave before barrier
#endif
    __syncthreads();

    // ---- Compute: software-pipelined fragment loads. ----
    v2f a0 = frag_rowmajor(As, wrow, 0, K1_LDA);
    v2f b0[4];
#pragma unroll
    for (int n = 0; n < 4; ++n) b0[n] = frag_rowmajor(Bs, n * 16, 0, K1_LDB);
#pragma unroll
    for (int k0 = 0; k0 < K1_TK; k0 += 4) {
      v2f a1 = a0;
      v2f b1[4];
      if (k0 + 4 < K1_TK) {
        a1 = frag_rowmajor(As, wrow, k0 + 4, K1_LDA);
#pragma unroll
        for (int n = 0; n < 4; ++n)
          b1[n] = frag_rowmajor(Bs, n * 16, k0 + 4, K1_LDB);
      } else {
#pragma unroll
        for (int n = 0; n < 4; ++n) b1[n] = b0[n];
      }
#pragma unroll
      for (int n = 0; n < 4; ++n) acc[n] = wmma_f32(a0, b0[n], acc[n]);
      a0 = a1;
#pragma unroll
      for (int n = 0; n < 4; ++n) b0[n] = b1[n];
    }
  }

  int rbase, cl;
  acc_rc(rbase, cl);
#pragma unroll
  for (int n = 0; n < 4; ++n) {
    const int col = n0 + n * 16 + cl;
    const float bv = bias[col];
#pragma unroll
    for (int i = 0; i < 8; ++i) {
      const int row = m0 + wrow + rbase + i;
      float v = acc[n][i] + bv;
      if (apply_phi) v = (v > 0.f) ? (v + 1.f) : __expf(v);  // elu(v)+1
      Y[row * DMODEL + col] = v;
    }
  }
}

// ---------------------------------------------------------------------------
// Kernel 2: k_reduced[c] = 1e-8 + sum_j phi_k[j][c].
// ---------------------------------------------------------------------------
__global__ __launch_bounds__(256) void ksum_kernel(const float* __restrict__ kh,
                                                   float* __restrict__ kred) {
  const int c = (int)(blockIdx.x * 256 + threadIdx.x);
  float s = 1e-8f;
  for (int j = 0; j < S_LEN; ++j) s += kh[j * DMODEL + c];
  kred[c] = s;
}

// ---------------------------------------------------------------------------
// Kernel 3: per-head chunked causal scan. One workgroup (8 waves) per head.
// State S (64x64) in LDS; chunk C=32:
//   out_c = (Q_c @ S_prev) + (mask(Q_c K_c^T) @ V_c),  out *= z,  S += K_c^T V_c
// ---------------------------------------------------------------------------
#define CHUNK 32
#define LDT 65     // stride for 64-wide tiles (odd -> conflict-free b32 reads)
#define LDSC 33    // stride for 32-wide score tile

__global__ __launch_bounds__(256) void linattn_kernel(
    const float* __restrict__ qh, const float* __restrict__ kh,
    const float* __restrict__ vh, const float* __restrict__ kred,
    float* __restrict__ out) {
  __shared__ float St[DHEAD * LDT];    // state [d][e]
  __shared__ float Qs[CHUNK * LDT];    // [l][d]
  __shared__ float Ks[CHUNK * LDT];    // [j][d]
  __shared__ float Vs[CHUNK * LDT];    // [j][e]
  __shared__ float Sc[CHUNK * LDSC];   // masked scores [l][j]
  __shared__ float zz[CHUNK];
  __shared__ float kr[DHEAD];

  const int h = (int)blockIdx.x;
  const int tid = (int)threadIdx.x;
  const int wave = tid >> 5;

  for (int i = tid; i < DHEAD * LDT; i += 256) St[i] = 0.f;
  if (tid < DHEAD) kr[tid] = kred[h * DHEAD + tid];
  __syncthreads();

  for (int c = 0; c < S_LEN / CHUNK; ++c) {
    const int s0 = c * CHUNK;
    // Load Q/K/V chunk (2048 floats each) straight into LDS.
    for (int i = tid; i < CHUNK * DHEAD; i += 256) {
      const int r = i >> 6;
      const int col = i & 63;
      const int g = (s0 + r) * DMODEL + h * DHEAD + col;
#if HAVE_ASYNC_LDS
      __builtin_amdgcn_global_load_async_to_lds_b32(
          GPTR(int, &qh[g]), LPTR(int, &Qs[r * LDT + col]), 0, 0);
      __builtin_amdgcn_global_load_async_to_lds_b32(
          GPTR(int, &kh[g]), LPTR(int, &Ks[r * LDT + col]), 0, 0);
      __builtin_amdgcn_global_load_async_to_lds_b32(
          GPTR(int, &vh[g]), LPTR(int, &Vs[r * LDT + col]), 0, 0);
#else
      Qs[r * LDT + col] = qh[g];
      Ks[r * LDT + col] = kh[g];
      Vs[r * LDT + col] = vh[g];
#endif
    }
#if HAVE_ASYNC_LDS
    __builtin_amdgcn_s_wait_asynccnt(0);
#endif
    __syncthreads();

    // z for this chunk (denominator uses the FULL key sum, per reference).
    if (tid < CHUNK) {
      float s = 0.f;
      for (int d = 0; d < DHEAD; ++d) s += Qs[tid * LDT + d] * kr[d];
      zz[tid] = 1.0f / s;
    }

    // Scores Q_c K_c^T: waves 0..3 cover the 32x32 block (wave-uniform branch,
    // EXEC stays all-ones inside the WMMA loop as required).
    v8f sc_acc = v8f{0, 0, 0, 0, 0, 0, 0, 0};
    const int stm = ((wave >> 1) & 1) * 16;
    const int stn = (wave & 1) * 16;
    if (wave < 4) {
      for (int k0 = 0; k0 < DHEAD; k0 += 4) {
        const v2f a = frag_rowmajor(Qs, stm, k0, LDT);   // A(l,d)
        const v2f b = frag_rowmajor(Ks, stn, k0, LDT);   // B(d,j)=Ks[j][d]
        sc_acc = wmma_f32(a, b, sc_acc);
      }
    }
    __syncthreads();
    if (wave < 4) {
      int rbase, cl;
      acc_rc(rbase, cl);
#pragma unroll
      for (int i = 0; i < 8; ++i) {
        const int r = stm + rbase + i;
        const int col = stn + cl;
        Sc[r * LDSC + col] = (col <= r) ? sc_acc[i] : 0.f;  // causal j<=l
      }
    }
    __syncthreads();

    // Output tile per wave: 32(l) x 64(e) = 8 tiles, one per wave.
    const int tm = (wave >> 2) * 16;   // l
    const int tn = (wave & 3) * 16;    // e
    v8f oacc = v8f{0, 0, 0, 0, 0, 0, 0, 0};
    // inter-chunk: Q_c @ S_prev
    for (int k0 = 0; k0 < DHEAD; k0 += 4) {
      const v2f a = frag_rowmajor(Qs, tm, k0, LDT);
      const v2f b = frag_colmajor(St, tn, k0, LDT);      // B(d,e)=St[d][e]
      oacc = wmma_f32(a, b, oacc);
    }
    // intra-chunk: masked scores @ V_c
    for (int k0 = 0; k0 < CHUNK; k0 += 4) {
      const v2f a = frag_rowmajor(Sc, tm, k0, LDSC);
      const v2f b = frag_colmajor(Vs, tn, k0, LDT);      // B(j,e)=Vs[j][e]
      oacc = wmma_f32(a, b, oacc);
    }
    // State delta K_c^T V_c: each wave owns 2 of the 16 (d x e) tiles.
    v8f kv[2];
    kv[0] = v8f{0, 0, 0, 0, 0, 0, 0, 0};
    kv[1] = v8f{0, 0, 0, 0, 0, 0, 0, 0};
#pragma unroll
    for (int t = 0; t < 2; ++t) {
      const int tt = wave * 2 + t;
      const int td = (tt >> 2) * 16;
      const int te = (tt & 3) * 16;
      for (int k0 = 0; k0 < CHUNK; k0 += 4) {
        const v2f a = frag_colmajor(Ks, td, k0, LDT);    // A(d,j)=Ks[j][d]
        const v2f b = frag_colmajor(Vs, te, k0, LDT);    // B(j,e)=Vs[j][e]
        kv[t] = wmma_f32(a, b, kv[t]);
      }
    }
    // Store scaled output.
    {
      int rbase, cl;
      acc_rc(rbase, cl);
#pragma unroll
      for (int i = 0; i < 8; ++i) {
        const int lr = tm + rbase + i;
        out[(s0 + lr) * DMODEL + h * DHEAD + tn + cl] = oacc[i] * zz[lr];
      }
    }
    __syncthreads();   // all reads of old St complete across waves
    // Commit S += K^T V (each element owned by exactly one lane).
    {
      int rbase, cl;
      acc_rc(rbase, cl);
#pragma unroll
      for (int t = 0; t < 2; ++t) {
        const int tt = wave * 2 + t;
        const int td = (tt >> 2) * 16;
        const int te = (tt & 3) * 16;
#pragma unroll
        for (int i = 0; i < 8; ++i)
          St[(td + rbase + i) * LDT + te + cl] += kv[t][i];
      }
    }
    __syncthreads();
  }
}

// ---------------------------------------------------------------------------
extern "C" void kernel_launch(void* const* d_in, const int* in_sizes, int n_in,
                              void* d_out, int out_size, void* d_ws,
                              size_t ws_size, hipStream_t stream) {
  (void)in_sizes; (void)n_in; (void)out_size; (void)ws_size;
  const float* q  = (const float*)d_in[0];
  const float* k  = (const float*)d_in[1];
  const float* v  = (const float*)d_in[2];
  // d_in[3] = mask: causality is hardcoded in linattn_kernel.
  const float* Wq = (const float*)d_in[4];
  const float* bq = (const float*)d_in[5];
  const float* Wk = (const float*)d_in[6];
  const float* bk = (const float*)d_in[7];
  const float* Wv = (const float*)d_in[8];
  const float* bv = (const float*)d_in[9];

  float* qh = (float*)d_ws;                    // (768,512) phi(q@Wq+bq)
  float* kh = qh + S_LEN * DMODEL;             // (768,512) phi(k@Wk+bk)
  float* vh = kh + S_LEN * DMODEL;             // (768,512) v@Wv+bv
  float* kr = vh + S_LEN * DMODEL;             // (512) column sums of kh
  float* out = (float*)d_out;                  // (768,512)

  const dim3 g1(DMODEL / K1_TN, S_LEN / K1_TM, 1);
  proj_phi_kernel<<<g1, 256, 0, stream>>>(q, Wq, bq, qh, 1);
  proj_phi_kernel<<<g1, 256, 0, stream>>>(k, Wk, bk, kh, 1);
  proj_phi_kernel<<<g1, 256, 0, stream>>>(v, Wv, bv, vh, 0);
  ksum_kernel<<<dim3(DMODEL / 256), 256, 0, stream>>>(kh, kr);
  linattn_kernel<<<dim3(NHEAD), 256, 0, stream>>>(qh, kh, vh, kr, out);
}